// SpeechTokenizer_87067577025303
// MI455X (gfx1250) — compile-verified
//
#include <hip/hip_runtime.h>
#include <hip/hip_bf16.h>

typedef __attribute__((ext_vector_type(2))) float v2f;
typedef __attribute__((ext_vector_type(8))) float v8f;
typedef __attribute__((ext_vector_type(4))) int   v4i;

#define DIM      512
#define KCODES   1024
#define NROWS    32000
#define ROWPAD   516            // 512 + 4 pad -> conflict-free LDS banks
#define WG_ROWS  128            // 8 waves * 16 rows
#define NTILES   (KCODES / 16)  // 64 column tiles

#define AS_GLOBAL __attribute__((address_space(1)))
#define AS_LDS    __attribute__((address_space(3)))

// ---- CDNA5 async global->LDS copy (ASYNCcnt path), 16B per lane ----------
__device__ __forceinline__ void async_ld16(void* lds, const void* g) {
#if __has_builtin(__builtin_amdgcn_global_load_async_to_lds_b128)
    __builtin_amdgcn_global_load_async_to_lds_b128(
        (AS_GLOBAL v4i*)g, (AS_LDS v4i*)lds, 0, 0);
#else
    asm volatile("global_load_async_to_lds_b128 %0, %1, off"
                 :: "v"((AS_LDS void*)lds), "v"(g)
                 : "memory");
#endif
}

__device__ __forceinline__ void wait_async0() {
#if __has_builtin(__builtin_amdgcn_s_wait_asynccnt)
    __builtin_amdgcn_s_wait_asynccnt(0);
#else
    asm volatile("s_wait_asynccnt 0x0" ::: "memory");
#endif
}

// ---------------------------------------------------------------------------
// Kernel 1: e_sq[k] = ||embed[k]||^2   (one wave32 per codeword)
// ---------------------------------------------------------------------------
__global__ void vq_esq_kernel(const float* __restrict__ embed,
                              float* __restrict__ esq) {
    const int k    = blockIdx.x * 8 + (threadIdx.x >> 5);
    const int lane = threadIdx.x & 31;
    const float* row = embed + (size_t)k * DIM;
    float s = 0.0f;
    #pragma unroll
    for (int i = 0; i < DIM / 32; ++i) {
        float v = row[lane + i * 32];
        s = fmaf(v, v, s);
    }
    #pragma unroll
    for (int off = 16; off > 0; off >>= 1)
        s += __shfl_xor(s, off, 32);
    if (lane == 0) esq[k] = s;
}

// ---------------------------------------------------------------------------
// Kernel 2: per-row argmax over codebook via fp32 WMMA, then gather.
//   score[n,k] = 2 * dot(x[n], e[k]) - ||e[k]||^2   (x-norm is row-constant)
// ---------------------------------------------------------------------------
__global__ void vq_argmax_kernel(const float* __restrict__ x,
                                 const float* __restrict__ embed,
                                 const float* __restrict__ esq,
                                 float* __restrict__ out) {
    extern __shared__ __align__(16) float smem[];
    float* sA = smem;                          // WG_ROWS * ROWPAD
    float* sB = smem + WG_ROWS * ROWPAD;       // 16 * ROWPAD
    float* sE = sB + 16 * ROWPAD;              // KCODES

    const int tid     = threadIdx.x;
    const int wave    = tid >> 5;
    const int lane    = tid & 31;
    const int sub     = lane & 15;             // N index / M index within frag
    const int kadd    = (lane >> 4) << 1;      // 0 for lanes 0-15, 2 for 16-31
    const int rowBase = blockIdx.x * WG_ROWS;

    // ---- async-stage A (128 rows of x), e_sq table, and B tile 0 ----
    {
        const float4* xg = (const float4*)(x + (size_t)rowBase * DIM);
        for (int i = tid; i < WG_ROWS * (DIM / 4); i += blockDim.x) {
            const int r = i >> 7;              // DIM/4 == 128 float4 per row
            const int c = i & 127;
            async_ld16(sA + r * ROWPAD + c * 4, xg + i);
        }
        async_ld16(sE + tid * 4, esq + tid * 4);      // 256 thr * 4 = 1024
        const float4* eg = (const float4*)embed;      // col tile 0
        for (int i = tid; i < 16 * (DIM / 4); i += blockDim.x) {
            const int r = i >> 7;
            const int c = i & 127;
            async_ld16(sB + r * ROWPAD + c * 4, eg + i);
        }
    }

    const float* aBase = sA + (wave * 16 + sub) * ROWPAD + kadd;
    const float* bBase = sB + sub * ROWPAD + kadd;

    float bestVal[8];
    int   bestIdx[8];
    #pragma unroll
    for (int j = 0; j < 8; ++j) { bestVal[j] = -3.4e38f; bestIdx[j] = 0; }

    for (int ct = 0; ct < NTILES; ++ct) {
        const int colBase = ct << 4;

        wait_async0();          // this wave's staged data has landed
        __syncthreads();        // every wave's staged data has landed

        // warm L2/WGP$ with next codebook tile while we compute this one
        if (ct + 1 < NTILES) {
            const char* p = (const char*)(embed + (size_t)(colBase + 16) * DIM);
            __builtin_prefetch(p + tid * 128, 0, 0);
        }

        // ---- 16x16 tile GEMM: 128 x v_wmma_f32_16x16x4_f32 ----
        v8f acc0 = {}, acc1 = {}, acc2 = {}, acc3 = {};
        #pragma unroll
        for (int k = 0; k < DIM; k += 16) {
            v2f a0 = *(const v2f*)(aBase + k);
            v2f b0 = *(const v2f*)(bBase + k);
            acc0 = __builtin_amdgcn_wmma_f32_16x16x4_f32(false, a0, false, b0,
                                                         (short)0, acc0, false, false);
            v2f a1 = *(const v2f*)(aBase + k + 4);
            v2f b1 = *(const v2f*)(bBase + k + 4);
            acc1 = __builtin_amdgcn_wmma_f32_16x16x4_f32(false, a1, false, b1,
                                                         (short)0, acc1, false, false);
            v2f a2 = *(const v2f*)(aBase + k + 8);
            v2f b2 = *(const v2f*)(bBase + k + 8);
            acc2 = __builtin_amdgcn_wmma_f32_16x16x4_f32(false, a2, false, b2,
                                                         (short)0, acc2, false, false);
            v2f a3 = *(const v2f*)(aBase + k + 12);
            v2f b3 = *(const v2f*)(bBase + k + 12);
            acc3 = __builtin_amdgcn_wmma_f32_16x16x4_f32(false, a3, false, b3,
                                                         (short)0, acc3, false, false);
        }

        const float es  = sE[colBase + sub];
        const int   col = colBase + sub;
        #pragma unroll
        for (int j = 0; j < 8; ++j) {
            const float dot = (acc0[j] + acc1[j]) + (acc2[j] + acc3[j]);
            const float v   = 2.0f * dot - es;
            if (v > bestVal[j]) { bestVal[j] = v; bestIdx[j] = col; }
        }

        __syncthreads();        // every wave done reading sB for this tile

        // ---- async-stage next B tile (single buffer, ASYNCcnt-tracked) ----
        if (ct + 1 < NTILES) {
            const float4* eg =
                (const float4*)(embed + (size_t)(colBase + 16) * DIM);
            for (int i = tid; i < 16 * (DIM / 4); i += blockDim.x) {
                const int r = i >> 7;
                const int c = i & 127;
                async_ld16(sB + r * ROWPAD + c * 4, eg + i);
            }
        }
    }

    // ---- reduce argmax across the 16 lanes of each half-wave group ----
    #pragma unroll
    for (int j = 0; j < 8; ++j) {
        float v  = bestVal[j];
        int   id = bestIdx[j];
        #pragma unroll
        for (int off = 8; off > 0; off >>= 1) {
            const float ov = __shfl_xor(v, off, 16);
            const int   oi = __shfl_xor(id, off, 16);
            if (ov > v || (ov == v && oi < id)) { v = ov; id = oi; }
        }
        bestIdx[j] = id;
    }

    // ---- write indices + gather quantized rows ----
    float* qOut = out;                               // NROWS * DIM floats
    float* iOut = out + (size_t)NROWS * DIM;         // NROWS indices (as float)
    const int myRowBase = rowBase + wave * 16;

    #pragma unroll
    for (int j = 0; j < 8; ++j) {
        // lane 0 holds result for row myRowBase+j; lane 16 for myRowBase+8+j
        const int idxLow  = __shfl(bestIdx[j], 0, 32);
        const int idxHigh = __shfl(bestIdx[j], 16, 32);
        if (lane == 0)  iOut[myRowBase + j]     = (float)idxLow;
        if (lane == 16) iOut[myRowBase + 8 + j] = (float)idxHigh;

        const float4* srcL = (const float4*)(embed + (size_t)idxLow * DIM);
        float4*       dstL = (float4*)(qOut + (size_t)(myRowBase + j) * DIM);
        #pragma unroll
        for (int c = 0; c < DIM / 4 / 32; ++c)
            dstL[lane + c * 32] = srcL[lane + c * 32];

        const float4* srcH = (const float4*)(embed + (size_t)idxHigh * DIM);
        float4*       dstH = (float4*)(qOut + (size_t)(myRowBase + 8 + j) * DIM);
        #pragma unroll
        for (int c = 0; c < DIM / 4 / 32; ++c)
            dstH[lane + c * 32] = srcH[lane + c * 32];
    }
}

// ---------------------------------------------------------------------------
extern "C" void kernel_launch(void* const* d_in, const int* in_sizes, int n_in,
                              void* d_out, int out_size, void* d_ws, size_t ws_size,
                              hipStream_t stream) {
    (void)in_sizes; (void)n_in; (void)out_size; (void)ws_size;
    const float* x     = (const float*)d_in[0];   // [16,2000,512] fp32
    const float* embed = (const float*)d_in[1];   // [1024,512]   fp32
    float* esq = (float*)d_ws;                    // 1024 floats scratch
    float* out = (float*)d_out;                   // quantize ++ indices

    vq_esq_kernel<<<KCODES / 8, 256, 0, stream>>>(embed, esq);

    const size_t shmem =
        (size_t)(WG_ROWS * ROWPAD + 16 * ROWPAD + KCODES) * sizeof(float);
    vq_argmax_kernel<<<NROWS / WG_ROWS, 256, shmem, stream>>>(x, embed, esq, out);
}